// FiltrationLayer_90821378441803
// MI455X (gfx1250) — compile-verified
//
#include <hip/hip_runtime.h>

#define ROW      4096
#define BLOCK    256
#define CHUNKS   4        // 4 float4 per thread: 4*4*256 = 4096 elements
#define NWAVES   (BLOCK / 32)
#define GRID_WG  2048

typedef __attribute__((ext_vector_type(4))) float v4f;   // clang vector: OK for NT builtins

// branchless top-2 insert: (m1,m2) <- top2 of {m1,m2,x}, invariant m1 >= m2
__device__ __forceinline__ void top2_insert(float x, float& m1, float& m2) {
    float nm1 = fmaxf(m1, x);
    m2 = fmaxf(m2, fminf(m1, x));
    m1 = nm1;
}

// merge two sorted pairs: (m1,m2) <- top2 of {m1,m2,b1,b2}
__device__ __forceinline__ void top2_merge(float b1, float b2, float& m1, float& m2) {
    float lo = fminf(m1, b1);
    m1 = fmaxf(m1, b1);
    m2 = fmaxf(lo, fmaxf(m2, b2));
}

// issue one row (16KB) as async global->LDS b128 copies: 4 per thread,
// each wave32 instruction moves 32 lanes * 16B = 512B, ASYNCcnt += 4 per wave.
// NT hint: row is read exactly once -> don't retain in L2 (192MB vs 512MB stream).
__device__ __forceinline__ void issue_row_async(const float* __restrict__ gbase,
                                                unsigned lds_base, int tid) {
    unsigned long long sbase = (unsigned long long)gbase;
#pragma unroll
    for (int c = 0; c < CHUNKS; ++c) {
        unsigned goff  = (unsigned)((c * 1024 + tid * 4) * 4);  // byte offset in row
        unsigned laddr = lds_base + goff;
        asm volatile("global_load_async_to_lds_b128 %0, %1, %2 th:TH_LOAD_NT"
                     :: "v"(laddr), "v"(goff), "s"(sbase)
                     : "memory");
    }
}

extern "C" __global__ __launch_bounds__(BLOCK)
void topk2_straight_through_kernel(const float* __restrict__ in,
                                   float* __restrict__ out,
                                   int nrows) {
    __shared__ float buf[2][ROW];      // 32 KB double buffer
    __shared__ float wpart1[NWAVES];
    __shared__ float wpart2[NWAVES];
    __shared__ float s_pivot;

    const int tid  = threadIdx.x;
    const int wid  = tid >> 5;
    const int lane = tid & 31;

    // LDS byte address = low 32 bits of the flat shared-aperture address
    const unsigned lds_b0 = (unsigned)(size_t)(&buf[0][0]);
    const unsigned lds_b1 = (unsigned)(size_t)(&buf[1][0]);

    int row = blockIdx.x;
    if (row >= nrows) return;

    // prologue: fill buffer 0 with the first row
    issue_row_async(in + (size_t)row * ROW, lds_b0, tid);

    int iter = 0;
    for (; row < nrows; row += gridDim.x, ++iter) {
        const int  nxt_row  = row + (int)gridDim.x;
        const bool has_next = nxt_row < nrows;
        const unsigned nxt_base = (iter & 1) ? lds_b0 : lds_b1;

        if (has_next) {
            // prefetch next row into the other buffer, then wait for only the
            // current buffer's 4 async loads (in-order completion per wave)
            issue_row_async(in + (size_t)nxt_row * ROW, nxt_base, tid);
            asm volatile("s_wait_asynccnt 4" ::: "memory");
        } else {
            asm volatile("s_wait_asynccnt 0" ::: "memory");
        }
        __syncthreads();   // all waves' portions of the current row are in LDS

        // pull 16 values into registers (ds_load_b128), local top-2
        const float* curf = (iter & 1) ? buf[1] : buf[0];
        v4f v[CHUNKS];
        float m1 = -__builtin_huge_valf();
        float m2 = -__builtin_huge_valf();
#pragma unroll
        for (int c = 0; c < CHUNKS; ++c) {
            v[c] = ((const v4f*)curf)[c * 256 + tid];
            top2_insert(v[c].x, m1, m2);
            top2_insert(v[c].y, m1, m2);
            top2_insert(v[c].z, m1, m2);
            top2_insert(v[c].w, m1, m2);
        }

        // wave32 butterfly reduction of the (m1,m2) pair
#pragma unroll
        for (int m = 16; m >= 1; m >>= 1) {
            float b1 = __shfl_xor(m1, m, 32);
            float b2 = __shfl_xor(m2, m, 32);
            top2_merge(b1, b2, m1, m2);
        }
        if (lane == 0) { wpart1[wid] = m1; wpart2[wid] = m2; }
        __syncthreads();   // also guarantees all LDS reads of cur buffer done

        if (tid == 0) {
            float r1 = wpart1[0], r2 = wpart2[0];
#pragma unroll
            for (int w = 1; w < NWAVES; ++w) top2_merge(wpart1[w], wpart2[w], r1, r2);
            s_pivot = r2;  // K-th largest (K=2)
        }
        __syncthreads();
        const float pivot = s_pivot;

        // straight-through forward: x < pivot -> 0 ; x >= pivot -> x + (1 - x)
        // (bit-exact match of: inputs - sg*mask + (1-sg)*(1-mask))
        // NT stores: written once, never re-read -> bypass L2 retention.
        v4f* orow = (v4f*)(out + (size_t)row * ROW);
#pragma unroll
        for (int c = 0; c < CHUNKS; ++c) {
            v4f o;
            o.x = (v[c].x >= pivot) ? (v[c].x + (1.0f - v[c].x)) : 0.0f;
            o.y = (v[c].y >= pivot) ? (v[c].y + (1.0f - v[c].y)) : 0.0f;
            o.z = (v[c].z >= pivot) ? (v[c].z + (1.0f - v[c].z)) : 0.0f;
            o.w = (v[c].w >= pivot) ? (v[c].w + (1.0f - v[c].w)) : 0.0f;
            __builtin_nontemporal_store(o, &orow[c * 256 + tid]);
        }
        // no trailing barrier needed: any wave entering the next iteration has
        // passed the wpart barrier, which is after every wave's LDS reads.
    }
}

extern "C" void kernel_launch(void* const* d_in, const int* in_sizes, int n_in,
                              void* d_out, int out_size, void* d_ws, size_t ws_size,
                              hipStream_t stream) {
    (void)n_in; (void)out_size; (void)d_ws; (void)ws_size;
    const float* in  = (const float*)d_in[0];
    float*       out = (float*)d_out;
    const int nrows = in_sizes[0] / ROW;            // 32768 for (16,2048,4096)
    const int grid  = nrows < GRID_WG ? nrows : GRID_WG;
    hipLaunchKernelGGL(topk2_straight_through_kernel,
                       dim3(grid), dim3(BLOCK), 0, stream, in, out, nrows);
}